// EnhancedFrequencyLogGaborConv2d_45243185496585
// MI455X (gfx1250) — compile-verified
//
#include <hip/hip_runtime.h>
#include <math.h>

// ---------------- problem constants ----------------
#define NN   384
#define PP   (NN * NN)          // 147456 plane
#define BB   4
#define CC   3                  // Cin == Cout == S == 3
#define NT16 (NN / 16)          // 24 tiles per 384 dim

typedef __attribute__((ext_vector_type(2))) float v2f;
typedef __attribute__((ext_vector_type(8))) float v8f;

// CDNA5 f32 WMMA: D(16x16) = A(16x4) * B(4x16) + C  -> v_wmma_f32_16x16x4_f32
__device__ __forceinline__ v8f wmma4(v2f a, v2f b, v8f c) {
  return __builtin_amdgcn_wmma_f32_16x16x4_f32(false, a, false, b, (short)0, c,
                                               false, false);
}

__device__ __forceinline__ int shift192(int i) { return i < 192 ? i + 192 : i - 192; }

// ---------------- 0: twiddle matrices ----------------
__global__ void trig_init(float* __restrict__ cosm, float* __restrict__ sinm) {
  int idx = blockIdx.x * blockDim.x + threadIdx.x;
  if (idx >= PP) return;
  int u = idx / NN, k = idx % NN;
  int m = (u * k) % NN;
  float ang = 6.283185307179586f * (float)m / (float)NN;
  cosm[idx] = cosf(ang);
  sinm[idx] = sinf(ang);
}

// ---------------- 1: forward row DFT, 64x16 per wave ----------------
// T[r, v] = sum_w x[r, w] * exp(-2pi i w v / N), r in [0,4608)
__global__ void fwd_rows(const float* __restrict__ x, const float* __restrict__ cosm,
                         const float* __restrict__ sinm, float* __restrict__ tre,
                         float* __restrict__ tim) {
  const int wave = threadIdx.x >> 5;
  const int lane = threadIdx.x & 31;
  const int tile = blockIdx.x * 4 + wave;      // 72 * 24 waves
  const int mt = tile / NT16, nt = tile % NT16;
  const int g = lane >> 4, ln = lane & 15;
  const int col = nt * 16 + ln;
  v8f accRe[4] = {}, accIm[4] = {};
  for (int k0 = 0; k0 < NN; k0 += 4) {
    const int ka = k0 + 2 * g;
    // prefetch twiddle stream 16 K-steps ahead (global_prefetch_b8)
    if (ka + 64 < NN) {
      __builtin_prefetch(&cosm[(ka + 64) * NN + col], 0, 3);
      __builtin_prefetch(&sinm[(ka + 64) * NN + col], 0, 3);
    }
    v2f bc; bc.x = cosm[ka * NN + col]; bc.y = cosm[(ka + 1) * NN + col];
    v2f bs; bs.x = sinm[ka * NN + col]; bs.y = sinm[(ka + 1) * NN + col];
#pragma unroll
    for (int s = 0; s < 4; ++s) {
      const int row = mt * 64 + s * 16 + ln;
      v2f a; a.x = x[row * NN + ka]; a.y = x[row * NN + ka + 1];
      accRe[s] = wmma4(a, bc, accRe[s]);
      accIm[s] = wmma4(a, bs, accIm[s]);
    }
  }
#pragma unroll
  for (int s = 0; s < 4; ++s)
    for (int r = 0; r < 8; ++r) {
      const int orow = mt * 64 + s * 16 + r + 8 * g;
      tre[orow * NN + col] = accRe[s][r];
      tim[orow * NN + col] = -accIm[s][r];
    }
}

// ---------------- 2: forward col DFT + full fftshift store, 32x16/wave ----------------
// X = (C - iS) * (Tre + iTim)
__global__ void fwd_cols(const float* __restrict__ cosm, const float* __restrict__ sinm,
                         const float* __restrict__ tre, const float* __restrict__ tim,
                         float* __restrict__ xsre, float* __restrict__ xsim) {
  const int wave = threadIdx.x >> 5;
  const int lane = threadIdx.x & 31;
  const int tile = blockIdx.x * 4 + wave;      // 12 * 12 * 24 waves
  const int plane = tile / (12 * NT16);
  const int t = tile % (12 * NT16);
  const int mt = t / NT16, nt = t % NT16;
  const int g = lane >> 4, ln = lane & 15;
  const int col = nt * 16 + ln;                // v
  const float* pre = tre + (size_t)plane * PP;
  const float* pim = tim + (size_t)plane * PP;
  v8f aCre[2] = {}, aSim[2] = {}, aCim[2] = {}, aSre[2] = {};
  for (int k0 = 0; k0 < NN; k0 += 4) {
    const int ka = k0 + 2 * g;
    v2f br; br.x = pre[ka * NN + col]; br.y = pre[(ka + 1) * NN + col];
    v2f bi; bi.x = pim[ka * NN + col]; bi.y = pim[(ka + 1) * NN + col];
#pragma unroll
    for (int s = 0; s < 2; ++s) {
      const int arow = mt * 32 + s * 16 + ln;  // u
      v2f ac; ac.x = cosm[arow * NN + ka]; ac.y = cosm[arow * NN + ka + 1];
      v2f as; as.x = sinm[arow * NN + ka]; as.y = sinm[arow * NN + ka + 1];
      aCre[s] = wmma4(ac, br, aCre[s]);
      aSim[s] = wmma4(as, bi, aSim[s]);
      aCim[s] = wmma4(ac, bi, aCim[s]);
      aSre[s] = wmma4(as, br, aSre[s]);
    }
  }
  const int b = plane / CC, c = plane % CC;
  const int bp = (b + 2) & 3, cp = (c + 1) % 3;
  const int vp = shift192(col);
#pragma unroll
  for (int s = 0; s < 2; ++s)
    for (int r = 0; r < 8; ++r) {
      const int u = mt * 32 + s * 16 + r + 8 * g;
      const size_t idx = (size_t)(bp * CC + cp) * PP + (size_t)shift192(u) * NN + vp;
      xsre[idx] = aCre[s][r] + aSim[s][r];     // Xre = C*Tre + S*Tim
      xsim[idx] = aCim[s][r] - aSre[s][r];     // Xim = C*Tim - S*Tre
    }
}

// ---------------- 3: fused attention MLP + log-Gabor + attended ----------------
__global__ void attn_attended(const float* __restrict__ xsre, const float* __restrict__ xsim,
                              const float* __restrict__ w1, const float* __restrict__ b1v,
                              const float* __restrict__ w2, const float* __restrict__ b2v,
                              const float* __restrict__ theta, const float* __restrict__ sigma,
                              const float* __restrict__ f0, const float* __restrict__ th0,
                              float* __restrict__ aure, float* __restrict__ auim) {
  const int idx = blockIdx.x * blockDim.x + threadIdx.x;
  if (idx >= BB * PP) return;
  const int b = idx / PP, pix = idx % PP;
  const int h = pix / NN, w = pix % NN;
  float xr[3], xi[3], mag[3];
  for (int d = 0; d < 3; ++d) {
    xr[d] = xsre[(b * CC + d) * PP + pix];
    xi[d] = xsim[(b * CC + d) * PP + pix];
    mag[d] = sqrtf(xr[d] * xr[d] + xi[d] * xi[d]);
  }
  float logits[9];
  for (int o = 0; o < 9; ++o) logits[o] = b2v[o];
  for (int j = 0; j < 64; ++j) {
    float hv = b1v[j] + w1[j * 3 + 0] * mag[0] + w1[j * 3 + 1] * mag[1] + w1[j * 3 + 2] * mag[2];
    hv = fmaxf(hv, 0.f);
    for (int o = 0; o < 9; ++o) logits[o] += w2[o * 64 + j] * hv;
  }
  float mx = logits[0];
  for (int o = 1; o < 9; ++o) mx = fmaxf(mx, logits[o]);
  float at[9], se = 0.f;
  for (int o = 0; o < 9; ++o) { at[o] = __expf(logits[o] - mx); se += at[o]; }
  const float inv = 1.f / se;
  const float yy = -1.f + 2.f * (float)h / 383.f;
  const float xx = -1.f + 2.f * (float)w / 383.f;
  const float r = sqrtf(xx * xx + yy * yy + 1e-6f);
  const float phi = atan2f(yy, xx);
  float wgt[3] = {0.f, 0.f, 0.f};
  for (int sc = 0; sc < 9; ++sc) {
    const float a = at[sc] * inv;
    for (int d = 0; d < 3; ++d) {
      const int pi = sc * 3 + d;
      const float la = logf(r - logf(f0[pi]));
      const float ls = logf(sigma[pi]);
      const float lg = __expf(-(la * la) / (2.f * ls * ls));
      const float dphi = phi - theta[pi];
      const float t0 = th0[pi];
      const float ang = __expf(-(dphi * dphi) / (2.f * t0 * t0));
      wgt[d] += a * lg * ang;
    }
  }
  float are = 0.f, aim = 0.f;
  for (int d = 0; d < 3; ++d) { are += xr[d] * wgt[d]; aim += xi[d] * wgt[d]; }
  const size_t o = (size_t)((b + 2) & 3) * PP + (size_t)shift192(h) * NN + shift192(w);
  aure[o] = are;
  auim[o] = aim;
}

// ---------------- 4: 3x3 spatial conv (zero pad) ----------------
__global__ void conv3x3(const float* __restrict__ x, const float* __restrict__ cw,
                        float* __restrict__ y) {
  const int idx = blockIdx.x * blockDim.x + threadIdx.x;
  if (idx >= BB * CC * PP) return;
  const int b = idx / (CC * PP);
  const int o = (idx / PP) % CC;
  const int pix = idx % PP;
  const int h = pix / NN, w = pix % NN;
  float s = 0.f;
  for (int i = 0; i < 3; ++i)
    for (int kh = 0; kh < 3; ++kh) {
      const int hh = h + kh - 1;
      if (hh < 0 || hh >= NN) continue;
      for (int kw = 0; kw < 3; ++kw) {
        const int ww = w + kw - 1;
        if (ww < 0 || ww >= NN) continue;
        s += x[((size_t)(b * CC + i)) * PP + hh * NN + ww] * cw[((o * CC + i) * 3 + kh) * 3 + kw];
      }
    }
  y[idx] = s;
}

// ---------------- 5: inverse row DFT (mask gated in A), 32x16/wave ----------------
// combo q: co=q/12, b2=(q/3)%4, cin=q%3 ; Z = bandmask*AU ; U = Z*(C+iS)
__global__ void inv_rows(const float* __restrict__ cosm, const float* __restrict__ sinm,
                         const float* __restrict__ aure, const float* __restrict__ auim,
                         const float* __restrict__ band, int batch,
                         float* __restrict__ ure, float* __restrict__ uim) {
  const int wave = threadIdx.x >> 5;
  const int lane = threadIdx.x & 31;
  const int tile = blockIdx.x * 4 + wave;      // 12 * 12 * 24 waves
  const int j = tile / (12 * NT16);
  const int q = batch * 12 + j;
  const int t = tile % (12 * NT16);
  const int mt = t / NT16, nt = t % NT16;
  const int g = lane >> 4, ln = lane & 15;
  const int co = q / 12, b2 = (q / 3) % 4, cin = q % 3;
  const int CO = (co + 1) % 3, CI = (cin + 1) % 3;
  const float bs0 = band[(CO * 3 + CI) * 2 + 0];
  const float bs1 = band[(CO * 3 + CI) * 2 + 1];
  const int sidx = (int)floorf((bs0 + 1.f) * 0.5f * (float)NN);
  const int eidx = (int)floorf((bs1 + 1.f) * 0.5f * (float)NN);
  const int col = nt * 16 + ln;
  const float* par = aure + (size_t)b2 * PP;
  const float* pai = auim + (size_t)b2 * PP;
  bool rowok[2];
  int urow[2];
#pragma unroll
  for (int s = 0; s < 2; ++s) {
    urow[s] = mt * 32 + s * 16 + ln;
    const int us = shift192(urow[s]);
    rowok[s] = (us >= sidx) && (us < eidx);
  }
  v8f ReC[2] = {}, ImS[2] = {}, ReS[2] = {}, ImC[2] = {};
  for (int k0 = 0; k0 < NN; k0 += 4) {
    const int ka = k0 + 2 * g;
    const int s0 = shift192(ka), s1 = shift192(ka + 1);
    const bool c0 = (s0 >= sidx) && (s0 < eidx);
    const bool c1 = (s1 >= sidx) && (s1 < eidx);
    v2f bc; bc.x = cosm[ka * NN + col]; bc.y = cosm[(ka + 1) * NN + col];
    v2f bsv; bsv.x = sinm[ka * NN + col]; bsv.y = sinm[(ka + 1) * NN + col];
#pragma unroll
    for (int s = 0; s < 2; ++s) {
      const bool k0ok = rowok[s] && c0, k1ok = rowok[s] && c1;
      v2f ar; ar.x = k0ok ? par[urow[s] * NN + ka] : 0.f;
      ar.y = k1ok ? par[urow[s] * NN + ka + 1] : 0.f;
      v2f ai; ai.x = k0ok ? pai[urow[s] * NN + ka] : 0.f;
      ai.y = k1ok ? pai[urow[s] * NN + ka + 1] : 0.f;
      ReC[s] = wmma4(ar, bc, ReC[s]);
      ImS[s] = wmma4(ai, bsv, ImS[s]);
      ReS[s] = wmma4(ar, bsv, ReS[s]);
      ImC[s] = wmma4(ai, bc, ImC[s]);
    }
  }
  float* pur = ure + (size_t)j * PP;
  float* pui = uim + (size_t)j * PP;
#pragma unroll
  for (int s = 0; s < 2; ++s)
    for (int r = 0; r < 8; ++r) {
      const int orow = mt * 32 + s * 16 + r + 8 * g;
      pur[orow * NN + col] = ReC[s][r] - ImS[s][r];   // Ure = Zre*C - Zim*S
      pui[orow * NN + col] = ReS[s][r] + ImC[s][r];   // Uim = Zre*S + Zim*C
    }
}

// ---------------- 6: inverse col DFT + mix + 4x b1 broadcast, 64x16/wave ----------------
// G = (C*Ure - S*Uim)/N^2 ; out = mix*G + (1-mix)*xsp[b2,cin]
__global__ void inv_cols(const float* __restrict__ cosm, const float* __restrict__ sinm,
                         const float* __restrict__ ure, const float* __restrict__ uim,
                         const float* __restrict__ xsp, const float* __restrict__ mixing,
                         int batch, float* __restrict__ out) {
  const int wave = threadIdx.x >> 5;
  const int lane = threadIdx.x & 31;
  const int tile = blockIdx.x * 4 + wave;      // 12 * 6 * 24 waves
  const int j = tile / (6 * NT16);
  const int q = batch * 12 + j;
  const int t = tile % (6 * NT16);
  const int mt = t / NT16, nt = t % NT16;
  const int g = lane >> 4, ln = lane & 15;
  const int co = q / 12, b2 = (q / 3) % 4, cin = q % 3;
  const int col = nt * 16 + ln;                // w
  const float* pur = ure + (size_t)j * PP;
  const float* pui = uim + (size_t)j * PP;
  v8f accC[4] = {}, accS[4] = {};
  for (int k0 = 0; k0 < NN; k0 += 4) {
    const int ka = k0 + 2 * g;
    v2f br; br.x = pur[ka * NN + col]; br.y = pur[(ka + 1) * NN + col];
    v2f bi; bi.x = pui[ka * NN + col]; bi.y = pui[(ka + 1) * NN + col];
#pragma unroll
    for (int s = 0; s < 4; ++s) {
      const int arow = mt * 64 + s * 16 + ln;  // h
      v2f ac; ac.x = cosm[arow * NN + ka]; ac.y = cosm[arow * NN + ka + 1];
      v2f as; as.x = sinm[arow * NN + ka]; as.y = sinm[arow * NN + ka + 1];
      accC[s] = wmma4(ac, br, accC[s]);
      accS[s] = wmma4(as, bi, accS[s]);
    }
  }
  const float mixv = mixing[0];
  const float invN2 = 1.f / (float)PP;
#pragma unroll
  for (int s = 0; s < 4; ++s)
    for (int r = 0; r < 8; ++r) {
      const int h = mt * 64 + s * 16 + r + 8 * g;
      const float gv = (accC[s][r] - accS[s][r]) * invN2;
      const float v =
          mixv * gv + (1.f - mixv) * xsp[(size_t)(b2 * CC + cin) * PP + h * NN + col];
      for (int b1 = 0; b1 < BB; ++b1) {
        const size_t oidx =
            ((((size_t)b1 * CC + co) * BB + b2) * CC + cin) * PP + (size_t)h * NN + col;
        out[oidx] = v;
      }
    }
}

extern "C" void kernel_launch(void* const* d_in, const int* in_sizes, int n_in,
                              void* d_out, int out_size, void* d_ws, size_t ws_size,
                              hipStream_t stream) {
  const float* x      = (const float*)d_in[0];
  const float* theta  = (const float*)d_in[1];
  const float* sigma  = (const float*)d_in[2];
  const float* f0     = (const float*)d_in[3];
  const float* th0    = (const float*)d_in[4];
  const float* w1     = (const float*)d_in[5];
  const float* b1v    = (const float*)d_in[6];
  const float* w2     = (const float*)d_in[7];
  const float* b2v    = (const float*)d_in[8];
  const float* convw  = (const float*)d_in[9];
  const float* mixing = (const float*)d_in[10];
  const float* band   = (const float*)d_in[11];
  float* out = (float*)d_out;

  if (ws_size < (size_t)70 * PP * sizeof(float)) return;  // ~41.3 MB scratch
  float* ws   = (float*)d_ws;
  float* cosm = ws;                 // P
  float* sinm = cosm + PP;          // P
  float* tre  = sinm + PP;          // 12P  (reused as Ure per batch)
  float* tim  = tre + 12 * PP;      // 12P  (reused as Uim per batch)
  float* xsre = tim + 12 * PP;      // 12P
  float* xsim = xsre + 12 * PP;     // 12P
  float* aure = xsim + 12 * PP;     // 4P
  float* auim = aure + 4 * PP;      // 4P
  float* xsp  = auim + 4 * PP;      // 12P

  trig_init<<<PP / 256, 256, 0, stream>>>(cosm, sinm);
  fwd_rows<<<432, 128, 0, stream>>>(x, cosm, sinm, tre, tim);            // 1728 waves
  fwd_cols<<<864, 128, 0, stream>>>(cosm, sinm, tre, tim, xsre, xsim);   // 3456 waves
  attn_attended<<<(BB * PP) / 256, 256, 0, stream>>>(xsre, xsim, w1, b1v, w2, b2v,
                                                     theta, sigma, f0, th0, aure, auim);
  conv3x3<<<(BB * CC * PP) / 256, 256, 0, stream>>>(x, convw, xsp);
  for (int batch = 0; batch < 3; ++batch) {
    inv_rows<<<864, 128, 0, stream>>>(cosm, sinm, aure, auim, band, batch, tre, tim);
    inv_cols<<<432, 128, 0, stream>>>(cosm, sinm, tre, tim, xsp, mixing, batch, out);
  }
}